// PINNModel_27556510171161
// MI455X (gfx1250) — compile-verified
//
#include <hip/hip_runtime.h>
#include <cstdint>

#define NUM_STEPS 4096
#define TPB 256
#define CHUNK (NUM_STEPS / TPB)      // 16 serial RK4 steps per thread
#define ROWS_PER_WG 32
#define DT 0.01f

typedef uint32_t u32x4 __attribute__((ext_vector_type(4)));
typedef uint32_t u32x8 __attribute__((ext_vector_type(8)));

struct Mat2 { float m00, m01, m10, m11; };

__device__ __forceinline__ Mat2 matmul(const Mat2& A, const Mat2& B) {
  Mat2 C;
  C.m00 = fmaf(A.m00, B.m00, A.m01 * B.m10);
  C.m01 = fmaf(A.m00, B.m01, A.m01 * B.m11);
  C.m10 = fmaf(A.m10, B.m00, A.m11 * B.m10);
  C.m11 = fmaf(A.m10, B.m01, A.m11 * B.m11);
  return C;
}

// f(s) = (vel, (-k*pos - c*vel)/m)  -- same arithmetic as reference
__device__ __forceinline__ float2 fEval(float2 s, float kk, float cc, float mm) {
  return make_float2(s.y, (-kk * s.x - cc * s.y) / mm);
}

__device__ __forceinline__ float2 rk4Step(float2 s, float kk, float cc, float mm) {
  float2 k1 = fEval(s, kk, cc, mm);
  float2 s2 = make_float2(s.x + 0.5f * DT * k1.x, s.y + 0.5f * DT * k1.y);
  float2 k2 = fEval(s2, kk, cc, mm);
  float2 s3 = make_float2(s.x + 0.5f * DT * k2.x, s.y + 0.5f * DT * k2.y);
  float2 k3 = fEval(s3, kk, cc, mm);
  float2 s4 = make_float2(s.x + DT * k3.x, s.y + DT * k3.y);
  float2 k4 = fEval(s4, kk, cc, mm);
  const float w = DT / 6.0f;
  return make_float2(s.x + w * (k1.x + 2.0f * k2.x + 2.0f * k3.x + k4.x),
                     s.y + w * (k1.y + 2.0f * k2.y + 2.0f * k3.y + k4.y));
}

__global__ __launch_bounds__(TPB) void pinn_rk4_broadcast(
    const float* __restrict__ m_p, const float* __restrict__ c_p,
    const float* __restrict__ k_p, const float* __restrict__ s0_p,
    float* __restrict__ out, int batch) {
  __shared__ float traj[NUM_STEPS];   // 16 KB trajectory, one per WG

  const float mm = m_p[0], cc = c_p[0], kk = k_p[0];
  const float2 s0 = make_float2(s0_p[0], s0_p[1]);
  const int tid = threadIdx.x;

  // ---- Phase 1: trajectory into LDS (parallel-in-time via linearity) ----
  float2 e0 = rk4Step(make_float2(1.0f, 0.0f), kk, cc, mm);
  float2 e1 = rk4Step(make_float2(0.0f, 1.0f), kk, cc, mm);
  Mat2 M = {e0.x, e1.x, e0.y, e1.y};

  // P = M^CHUNK (CHUNK=16 -> 4 squarings)
  Mat2 P = M;
  #pragma unroll
  for (int i = 0; i < 4; ++i) P = matmul(P, P);

  // Q = P^tid (binary exponentiation, tid < 256)
  Mat2 Q = {1.0f, 0.0f, 0.0f, 1.0f};
  {
    unsigned e = (unsigned)tid;
    Mat2 B = P;
    while (e) {
      if (e & 1u) Q = matmul(Q, B);
      B = matmul(B, B);
      e >>= 1u;
    }
  }

  // Checkpoint s_{CHUNK*tid} = Q * s0, then CHUNK faithful serial steps.
  float2 s = make_float2(fmaf(Q.m00, s0.x, Q.m01 * s0.y),
                         fmaf(Q.m10, s0.x, Q.m11 * s0.y));
  const int base = tid * CHUNK;
  #pragma unroll
  for (int i = 0; i < CHUNK; ++i) {
    s = rk4Step(s, kk, cc, mm);
    traj[base + i] = s.x;            // out[t] = pos after step t+1
  }
  __syncthreads();

  // ---- Phase 2: TDM broadcast — one tensor_store_from_lds per output row ----
  // TDM ignores EXEC and issues per wave: restrict to wave 0 of the WG.
  if (tid < 32) {
    const unsigned lds0 = (unsigned)(uintptr_t)&traj[0];
    const int row0 = blockIdx.x * ROWS_PER_WG;
    const uint64_t obase = (uint64_t)(out + (size_t)row0 * NUM_STEPS);

    // D# group1: workgroup_mask=0; data_size=2 (4B); no barrier/iterate/pad.
    // tensor_dim0=4096 @bit48, tensor_dim1=1 @bit80, tile_dim0=4096 @bit112,
    // tile_dim1=1 @bit128, tile_dim2=0, tensor_dim0_stride=4096 @bit160.
    u32x8 g1;
    g1[0] = 0x00020000u;  // data_size=2 (4 bytes)
    g1[1] = 0x10000000u;  // tensor_dim0 = 4096 (low16 in [31:16])
    g1[2] = 0x00010000u;  // tensor_dim0 hi=0, tensor_dim1 = 1
    g1[3] = 0x10000000u;  // tensor_dim1 hi=0, tile_dim0 = 4096
    g1[4] = 0x00000001u;  // tile_dim1 = 1, tile_dim2 = 0 (unused)
    g1[5] = 4096u;        // tensor_dim0_stride lo
    g1[6] = 0u;           // stride0 hi, tensor_dim1_stride lo16
    g1[7] = 0u;           // tensor_dim1_stride hi
    // Groups 2/3: benign (tensor dims=1, tile_dim3/4=0 -> unused).
    const u32x4 g2 = {1u, 1u, 0u, 0u};            // tensor_dim2=1, tensor_dim3=1
    const u32x4 g3 = {0u, 0x00010000u, 0u, 0u};   // tensor_dim4=1

    for (int r = 0; r < ROWS_PER_WG; ++r) {
      const int row = row0 + r;
      if (row >= batch) break;
      const uint64_t ga = obase + (uint64_t)r * (NUM_STEPS * 4u);
      u32x4 g0;
      g0[0] = 1u;                    // count=1 (valid), not restore, no gather
      g0[1] = lds0;                  // lds_addr (bytes)
      g0[2] = (uint32_t)ga;          // global_addr[31:0]
      g0[3] = ((uint32_t)(ga >> 32) & 0x01FFFFFFu) | (2u << 30); // addr[56:32]|type=2
      asm volatile("tensor_store_from_lds %0, %1, %2, %3"
                   :: "s"(g0), "s"(g1), "s"(g2), "s"(g3)
                   : "memory");
    }
    __builtin_amdgcn_s_wait_tensorcnt(0);
  }
}

extern "C" void kernel_launch(void* const* d_in, const int* in_sizes, int n_in,
                              void* d_out, int out_size, void* d_ws, size_t ws_size,
                              hipStream_t stream) {
  // Inputs (setup_inputs order): x (unused by dynamics), m, c, k, initial_state
  const float* m_p  = (const float*)d_in[1];
  const float* c_p  = (const float*)d_in[2];
  const float* k_p  = (const float*)d_in[3];
  const float* s0_p = (const float*)d_in[4];
  float* out = (float*)d_out;

  const int batch = out_size / NUM_STEPS;                    // 16384
  const int grid = (batch + ROWS_PER_WG - 1) / ROWS_PER_WG;  // 512
  pinn_rk4_broadcast<<<grid, TPB, 0, stream>>>(m_p, c_p, k_p, s0_p, out, batch);
}